// AgentAwareAttention_901943132441
// MI455X (gfx1250) — compile-verified
//
#include <hip/hip_runtime.h>

// ---------------------------------------------------------------------------
// AgentAwareAttention on MI455X (gfx1250, wave32)
// All GEMMs on the matrix pipe via V_WMMA_F32_16X16X4_F32 (exact fp32, RNE).
// Edge handling: M/N-tail fragments use clamped addresses (their garbage only
// reaches C rows/cols that are never stored) -> branchless main loops with no
// exec-mask save/restore. Only the K-tail of attn@V needs a guarded step.
// ---------------------------------------------------------------------------

typedef float v2f __attribute__((ext_vector_type(2)));
typedef float v8f __attribute__((ext_vector_type(8)));

constexpr int BATCH   = 4;
constexpr int SEQ     = 2009;          // N
constexpr int DIM     = 512;
constexpr int HEADS   = 8;
constexpr int DHEAD   = 64;
constexpr int BH      = BATCH * HEADS;           // 32
constexpr int MROWS   = BATCH * SEQ;             // 8036
constexpr int LB      = 10;                      // look_back
constexpr int AGENT_N = 2000;                    // masked rows/cols
constexpr float SCALE = 0.125f;                  // 64^-0.5
constexpr size_t SLAB = (size_t)BATCH * HEADS * SEQ * DHEAD;  // 4,114,432
constexpr int CTILES  = (SEQ + 15) / 16;         // 126
constexpr int KMAIN   = (SEQ - 1) & ~3;          // 2008: unguarded K range

__device__ inline v8f zero_v8f() {
  v8f z;
#pragma unroll
  for (int i = 0; i < 8; ++i) z[i] = 0.0f;
  return z;
}

__device__ inline v8f wmma_f32(v2f a, v2f b, v8f c) {
  return __builtin_amdgcn_wmma_f32_16x16x4_f32(
      false, a, false, b, (short)0, c, false, false);
}

__device__ inline int imin(int a, int b) { return a < b ? a : b; }

// ---------------------------------------------------------------------------
// Kernel 1: fused QKV + QK_self projection, 32x64 tile per wave.
//   [8036 x 512] @ [512 x 2560] -> q/k/v/qs/ks slabs [B,H,N,64] in ws.
// ---------------------------------------------------------------------------
__global__ __launch_bounds__(32) void proj_kernel(
    const float* __restrict__ x, const float* __restrict__ wqkv,
    const float* __restrict__ wqks, float* __restrict__ ws) {
  const int lane = threadIdx.x & 31;
  const int lh = lane >> 4, lm = lane & 15;
  const int cbase = blockIdx.x * 64;   // 64-tiles never straddle col 1536
  const int rbase = blockIdx.y * 32;

  const float* wmat;
  int wld, wc0;
  if (cbase < 1536) { wmat = wqkv; wld = 1536; wc0 = cbase; }
  else              { wmat = wqks; wld = 1024; wc0 = cbase - 1536; }

  v8f acc[2][4];
#pragma unroll
  for (int rb = 0; rb < 2; ++rb)
#pragma unroll
    for (int s = 0; s < 4; ++s) acc[rb][s] = zero_v8f();

  // Clamped row addresses: out-of-range rows produce garbage only in C rows
  // that are never stored.
  const float* xr0 = x + (size_t)imin(rbase + lm, MROWS - 1) * DIM;
  const float* xr1 = x + (size_t)imin(rbase + 16 + lm, MROWS - 1) * DIM;

  for (int k0 = 0; k0 < DIM; k0 += 4) {
    const int ka = k0 + 2 * lh;
    const v2f a0 = *(const v2f*)(xr0 + ka);
    const v2f a1 = *(const v2f*)(xr1 + ka);
    const float* wr0 = wmat + (size_t)ka * wld + wc0;
    const float* wr1 = wr0 + wld;
#pragma unroll
    for (int s = 0; s < 4; ++s) {
      v2f b;
      b.x = wr0[s * 16 + lm];
      b.y = wr1[s * 16 + lm];
      acc[0][s] = wmma_f32(a0, b, acc[0][s]);
      acc[1][s] = wmma_f32(a1, b, acc[1][s]);
    }
  }

#pragma unroll
  for (int s = 0; s < 4; ++s) {
    const int col = cbase + s * 16 + lm;
    const int t  = (col < 1536) ? (col >> 9) : (3 + ((col - 1536) >> 9));
    const int cc = (col < 1536) ? (col & 511) : ((col - 1536) & 511);
    const int h = cc >> 6, d = cc & 63;
    float* base = ws + (size_t)t * SLAB;
#pragma unroll
    for (int rb = 0; rb < 2; ++rb)
#pragma unroll
      for (int r = 0; r < 8; ++r) {
        const int gr = rbase + rb * 16 + r + 8 * lh;
        if (gr < MROWS) {
          const int b = gr / SEQ, n = gr % SEQ;
          base[(((size_t)(b * HEADS + h)) * SEQ + n) * DHEAD + d] =
              acc[rb][s][r];
        }
      }
  }
}

// ---------------------------------------------------------------------------
// Kernel 2: blended pre-softmax scores, 32 rows per wave.
// q/qs fragments register-resident (128 VGPRs); per K-step only 2 v2f loads
// feed 4 WMMAs. Column tails use clamped addresses (unstored C columns).
// ---------------------------------------------------------------------------
__global__ __launch_bounds__(32) void scores_kernel(
    const float* __restrict__ ws, float* __restrict__ attn) {
  const int lane = threadIdx.x & 31;
  const int lh = lane >> 4, lm = lane & 15;
  const int bh = blockIdx.y;
  const int rbase = blockIdx.x * 32;

  const float* qb  = ws + 0 * SLAB + (size_t)bh * SEQ * DHEAD;
  const float* kb  = ws + 1 * SLAB + (size_t)bh * SEQ * DHEAD;
  const float* qsb = ws + 3 * SLAB + (size_t)bh * SEQ * DHEAD;
  const float* ksb = ws + 4 * SLAB + (size_t)bh * SEQ * DHEAD;

  v2f qa0[16], qa1[16], qsa0[16], qsa1[16];
  {
    const float* q0  = qb  + (size_t)imin(rbase + lm, SEQ - 1) * DHEAD;
    const float* q1  = qb  + (size_t)imin(rbase + 16 + lm, SEQ - 1) * DHEAD;
    const float* qs0 = qsb + (size_t)imin(rbase + lm, SEQ - 1) * DHEAD;
    const float* qs1 = qsb + (size_t)imin(rbase + 16 + lm, SEQ - 1) * DHEAD;
#pragma unroll
    for (int s = 0; s < 16; ++s) {
      const int d = s * 4 + 2 * lh;
      qa0[s]  = *(const v2f*)(q0 + d);
      qa1[s]  = *(const v2f*)(q1 + d);
      qsa0[s] = *(const v2f*)(qs0 + d);
      qsa1[s] = *(const v2f*)(qs1 + d);
    }
  }

  float* ab = attn + (size_t)bh * SEQ * SEQ;

  for (int ct = 0; ct < CTILES; ++ct) {
    const int m = ct * 16 + lm;
    const bool mv = m < SEQ;
    const int mc = imin(m, SEQ - 1);
    const float* krow  = kb  + (size_t)mc * DHEAD;
    const float* ksrow = ksb + (size_t)mc * DHEAD;

    v8f c0 = zero_v8f(), c1 = zero_v8f();
    v8f cs0 = zero_v8f(), cs1 = zero_v8f();
#pragma unroll
    for (int s = 0; s < 16; ++s) {
      const int d = s * 4 + 2 * lh;
      const v2f b1 = *(const v2f*)(krow + d);
      const v2f b2 = *(const v2f*)(ksrow + d);
      c0  = wmma_f32(qa0[s],  b1, c0);
      c1  = wmma_f32(qa1[s],  b1, c1);
      cs0 = wmma_f32(qsa0[s], b2, cs0);
      cs1 = wmma_f32(qsa1[s], b2, cs1);
    }

    const int magent = m / LB;
    const bool mblk = mv && (m < AGENT_N);
#pragma unroll
    for (int rb = 0; rb < 2; ++rb) {
      const v8f& cd = rb ? c1 : c0;
      const v8f& cs = rb ? cs1 : cs0;
#pragma unroll
      for (int r = 0; r < 8; ++r) {
        const int n = rbase + rb * 16 + r + 8 * lh;
        if (mv && n < SEQ) {
          const bool same = mblk && (n < AGENT_N) && ((n / LB) == magent);
          ab[(size_t)n * SEQ + m] = SCALE * (same ? cs[r] : cd[r]);
        }
      }
    }
  }
}

// ---------------------------------------------------------------------------
// Kernel 3: in-place row softmax over 2009 columns (256 thr/row, wave32
// shfl_xor + LDS cross-wave reduction).
// ---------------------------------------------------------------------------
__global__ __launch_bounds__(256) void softmax_kernel(float* __restrict__ attn) {
  const int row = blockIdx.x;
  const int bh = blockIdx.y;
  float* a = attn + ((size_t)bh * SEQ + row) * (size_t)SEQ;
  const int t = threadIdx.x;
  const int wid = t >> 5, ln = t & 31;

  __shared__ float redmax[8], redsum[8], bcast[2];

  float v[8];
  float mx = -3.4028235e38f;
#pragma unroll
  for (int j = 0; j < 8; ++j) {
    const int m = t + j * 256;
    v[j] = (m < SEQ) ? a[m] : -3.4028235e38f;
    mx = fmaxf(mx, v[j]);
  }
#pragma unroll
  for (int o = 16; o > 0; o >>= 1) mx = fmaxf(mx, __shfl_xor(mx, o, 32));
  if (ln == 0) redmax[wid] = mx;
  __syncthreads();
  if (t == 0) {
    float m2 = redmax[0];
    for (int i = 1; i < 8; ++i) m2 = fmaxf(m2, redmax[i]);
    bcast[0] = m2;
  }
  __syncthreads();
  const float bm = bcast[0];

  float sum = 0.0f;
#pragma unroll
  for (int j = 0; j < 8; ++j) {
    const int m = t + j * 256;
    v[j] = (m < SEQ) ? __expf(v[j] - bm) : 0.0f;
    sum += v[j];
  }
#pragma unroll
  for (int o = 16; o > 0; o >>= 1) sum += __shfl_xor(sum, o, 32);
  if (ln == 0) redsum[wid] = sum;
  __syncthreads();
  if (t == 0) {
    float s2 = 0.0f;
    for (int i = 0; i < 8; ++i) s2 += redsum[i];
    bcast[1] = 1.0f / s2;
  }
  __syncthreads();
  const float inv = bcast[1];
#pragma unroll
  for (int j = 0; j < 8; ++j) {
    const int m = t + j * 256;
    if (m < SEQ) a[m] = v[j] * inv;
  }
}

// ---------------------------------------------------------------------------
// Kernel 4: o = attn @ v (32x64 tile/wave, K=2009). Main K loop is fully
// unguarded; single guarded tail step uses clamped loads + selects (no exec
// branching). Output merged to [8036 x 512].
// ---------------------------------------------------------------------------
__global__ __launch_bounds__(32) void av_kernel(
    const float* __restrict__ attn, const float* __restrict__ vws,
    float* __restrict__ o) {
  const int lane = threadIdx.x & 31;
  const int lh = lane >> 4, lm = lane & 15;
  const int bh = blockIdx.y;
  const int b = bh >> 3, h = bh & 7;
  const int rbase = blockIdx.x * 32;

  const float* ab = attn + (size_t)bh * SEQ * SEQ;
  const float* vb = vws + (size_t)bh * SEQ * DHEAD;

  v8f acc[2][4];
#pragma unroll
  for (int rb = 0; rb < 2; ++rb)
#pragma unroll
    for (int s = 0; s < 4; ++s) acc[rb][s] = zero_v8f();

  const float* ar0 = ab + (size_t)imin(rbase + lm, SEQ - 1) * SEQ;
  const float* ar1 = ab + (size_t)imin(rbase + 16 + lm, SEQ - 1) * SEQ;

  for (int k0 = 0; k0 < KMAIN; k0 += 4) {   // ka max = 2007 < SEQ
    const int ka = k0 + 2 * lh;
    v2f a0, a1;
    a0.x = ar0[ka]; a0.y = ar0[ka + 1];     // stride-2009 rows: b32 pairs
    a1.x = ar1[ka]; a1.y = ar1[ka + 1];
    const float* vr0 = vb + (size_t)ka * DHEAD;
    const float* vr1 = vr0 + DHEAD;
#pragma unroll
    for (int s = 0; s < 4; ++s) {
      v2f bf;
      bf.x = vr0[s * 16 + lm];
      bf.y = vr1[s * 16 + lm];
      acc[0][s] = wmma_f32(a0, bf, acc[0][s]);
      acc[1][s] = wmma_f32(a1, bf, acc[1][s]);
    }
  }
  {  // K tail: k = 2008..2011, zero A beyond SEQ (branchless selects)
    const int ka = KMAIN + 2 * lh;
    const int kc0 = imin(ka, SEQ - 1), kc1 = imin(ka + 1, SEQ - 1);
    const bool v0 = ka < SEQ, v1 = (ka + 1) < SEQ;
    v2f a0, a1;
    const float t00 = ar0[kc0], t01 = ar0[kc1];
    const float t10 = ar1[kc0], t11 = ar1[kc1];
    a0.x = v0 ? t00 : 0.0f; a0.y = v1 ? t01 : 0.0f;
    a1.x = v0 ? t10 : 0.0f; a1.y = v1 ? t11 : 0.0f;
    const float* vr0 = vb + (size_t)kc0 * DHEAD;
    const float* vr1 = vb + (size_t)kc1 * DHEAD;
#pragma unroll
    for (int s = 0; s < 4; ++s) {
      v2f bf;
      bf.x = vr0[s * 16 + lm];
      bf.y = vr1[s * 16 + lm];
      acc[0][s] = wmma_f32(a0, bf, acc[0][s]);
      acc[1][s] = wmma_f32(a1, bf, acc[1][s]);
    }
  }

#pragma unroll
  for (int s = 0; s < 4; ++s) {
    const int dcol = h * 64 + s * 16 + lm;
#pragma unroll
    for (int rb = 0; rb < 2; ++rb)
#pragma unroll
      for (int r = 0; r < 8; ++r) {
        const int n = rbase + rb * 16 + r + 8 * lh;
        if (n < SEQ)
          o[((size_t)(b * SEQ + n)) * DIM + dcol] = acc[rb][s][r];
      }
  }
}

// ---------------------------------------------------------------------------
// Kernel 5: out = o @ w_out + b_out (32x64 tile/wave).
// ---------------------------------------------------------------------------
__global__ __launch_bounds__(32) void outproj_kernel(
    const float* __restrict__ o, const float* __restrict__ w,
    const float* __restrict__ bias, float* __restrict__ out) {
  const int lane = threadIdx.x & 31;
  const int lh = lane >> 4, lm = lane & 15;
  const int cbase = blockIdx.x * 64;
  const int rbase = blockIdx.y * 32;

  v8f acc[2][4];
#pragma unroll
  for (int rb = 0; rb < 2; ++rb)
#pragma unroll
    for (int s = 0; s < 4; ++s) acc[rb][s] = zero_v8f();

  const float* or0 = o + (size_t)imin(rbase + lm, MROWS - 1) * DIM;
  const float* or1 = o + (size_t)imin(rbase + 16 + lm, MROWS - 1) * DIM;

  for (int k0 = 0; k0 < DIM; k0 += 4) {
    const int ka = k0 + 2 * lh;
    const v2f a0 = *(const v2f*)(or0 + ka);
    const v2f a1 = *(const v2f*)(or1 + ka);
    const float* wr0 = w + (size_t)ka * DIM + cbase;
    const float* wr1 = wr0 + DIM;
#pragma unroll
    for (int s = 0; s < 4; ++s) {
      v2f b;
      b.x = wr0[s * 16 + lm];
      b.y = wr1[s * 16 + lm];
      acc[0][s] = wmma_f32(a0, b, acc[0][s]);
      acc[1][s] = wmma_f32(a1, b, acc[1][s]);
    }
  }

#pragma unroll
  for (int s = 0; s < 4; ++s) {
    const int col = cbase + s * 16 + lm;
    const float bb = bias[col];
#pragma unroll
    for (int rb = 0; rb < 2; ++rb)
#pragma unroll
      for (int r = 0; r < 8; ++r) {
        const int gr = rbase + rb * 16 + r + 8 * lh;
        if (gr < MROWS) out[(size_t)gr * DIM + col] = acc[rb][s][r] + bb;
      }
  }
}

// ---------------------------------------------------------------------------
// Host launcher. d_out = [out (4*2009*512)] ++ [attn (4*8*2009*2009)].
// d_ws: 6 slabs of SLAB floats: q,k,v,qs,ks,o_merged (~99 MB).
// ---------------------------------------------------------------------------
extern "C" void kernel_launch(void* const* d_in, const int* in_sizes, int n_in,
                              void* d_out, int out_size, void* d_ws,
                              size_t ws_size, hipStream_t stream) {
  (void)in_sizes; (void)n_in; (void)out_size; (void)ws_size;
  const float* x    = (const float*)d_in[0];
  const float* wqkv = (const float*)d_in[1];
  const float* wqks = (const float*)d_in[2];
  const float* wout = (const float*)d_in[3];
  const float* bout = (const float*)d_in[4];

  float* out  = (float*)d_out;
  float* attn = out + (size_t)BATCH * SEQ * DIM;
  float* ws   = (float*)d_ws;
  float* o_ws = ws + 5 * SLAB;

  const int rtiles32  = (MROWS + 31) / 32;  // 252
  const int seqtiles32 = (SEQ + 31) / 32;   // 63

  dim3 g1(2560 / 64, rtiles32);
  proj_kernel<<<g1, 32, 0, stream>>>(x, wqkv, wqks, ws);

  dim3 g2(seqtiles32, BH);
  scores_kernel<<<g2, 32, 0, stream>>>(ws, attn);

  dim3 g3(SEQ, BH);
  softmax_kernel<<<g3, 256, 0, stream>>>(attn);

  av_kernel<<<g2, 32, 0, stream>>>(attn, ws + 2 * SLAB, o_ws);

  dim3 g5(DIM / 64, rtiles32);
  outproj_kernel<<<g5, 32, 0, stream>>>(o_ws, wout, bout, out);
}